// StVKSprings_10230612099764
// MI455X (gfx1250) — compile-verified
//
#include <hip/hip_runtime.h>

// StVK spring energy: e = 0.5 * sum_e k_e * (|x[i0]-x[i1]|^2 - l0_e^2)^2
// Memory-bound (~140 MB HBM vs ~0.1 GFLOP). Design: wide coalesced streams,
// L2-resident vertex gathers, fp32 accumulation, deterministic 2-pass
// reduction. Wave-level reduction runs on the CDNA5 matrix pipe via
// V_WMMA_F32_16X16X4_F32 (A = lane partials, B = ones -> row sums).

typedef __attribute__((ext_vector_type(2))) float v2f;
typedef __attribute__((ext_vector_type(8))) float v8f;

#define TPB 256
#define MAX_BLOCKS 1024

// Full-wave fp32 sum using the matrix unit. Must be called with EXEC all-1s
// (all 32 lanes converged). Layout (16x4 f32 A-matrix, wave32):
//   lanes 0-15 : VGPR0 = A[lane][0],    VGPR1 = A[lane][1]
//   lanes 16-31: VGPR0 = A[lane-16][2], VGPR1 = A[lane-16][3]
// With B = ones (4x16), D[m][n] = sum_k A[m][k] = s(m) + s(m+16).
// Each lane's 8 D-VGPRs are D[j][n] (lanes 0-15, m=0..7) or D[j+8][n]
// (lanes 16-31, m=8..15); summing them + xor-16 shuffle yields the wave sum.
__device__ __forceinline__ float wave_sum_wmma(float s) {
    v2f a; a[0] = s;    a[1] = 0.0f;   // covers all 32 lane values, rest zero
    v2f b; b[0] = 1.0f; b[1] = 1.0f;   // 4x16 ones
    v8f c = {};
    c = __builtin_amdgcn_wmma_f32_16x16x4_f32(
            /*neg_a=*/false, a, /*neg_b=*/false, b,
            /*c_mod=*/(short)0, c, /*reuse_a=*/false, /*reuse_b=*/false);
    float t = c[0] + c[1] + c[2] + c[3] + c[4] + c[5] + c[6] + c[7];
    t += __shfl_xor(t, 16);            // combine the two half-wave row groups
    return t;
}

__device__ __forceinline__ float block_sum(float acc, float* smem) {
    float w = wave_sum_wmma(acc);
    const int lane = threadIdx.x & 31;
    const int wid  = threadIdx.x >> 5;
    if (lane == 0) smem[wid] = w;
    __syncthreads();
    float total = 0.0f;
    if (threadIdx.x == 0) {
        #pragma unroll
        for (int i = 0; i < TPB / 32; ++i) total += smem[i];
    }
    return total;   // valid in thread 0 only
}

__global__ __launch_bounds__(TPB)
void stvk_partial_kernel(const float* __restrict__ x,
                         const float* __restrict__ l0,
                         const float* __restrict__ ks,
                         const int*   __restrict__ idx,   // [E][2] int32
                         float* __restrict__ partial,
                         int E) {
    const int tid    = blockIdx.x * TPB + threadIdx.x;
    const int stride = gridDim.x * TPB;
    const int npairs = E >> 1;

    const int4*   idx4 = (const int4*)idx;   // two edges per 16B load
    const float2* l02  = (const float2*)l0;
    const float2* ks2  = (const float2*)ks;

    float acc = 0.0f;

    for (int p = tid; p < npairs; p += stride) {
        const int4   ii = idx4[p];
        const float2 l  = l02[p];
        const float2 kk = ks2[p];

        // edge 0: gathers hit L2 (x table is 12 MB << 192 MB L2)
        {
            const float* xa = x + 3u * (unsigned)ii.x;
            const float* xb = x + 3u * (unsigned)ii.y;
            const float dx = xa[0] - xb[0];
            const float dy = xa[1] - xb[1];
            const float dz = xa[2] - xb[2];
            const float q  = fmaf(dz, dz, fmaf(dy, dy, dx * dx));
            const float dq = fmaf(-l.x, l.x, q);
            acc = fmaf(kk.x * dq, dq, acc);
        }
        // edge 1
        {
            const float* xa = x + 3u * (unsigned)ii.z;
            const float* xb = x + 3u * (unsigned)ii.w;
            const float dx = xa[0] - xb[0];
            const float dy = xa[1] - xb[1];
            const float dz = xa[2] - xb[2];
            const float q  = fmaf(dz, dz, fmaf(dy, dy, dx * dx));
            const float dq = fmaf(-l.y, l.y, q);
            acc = fmaf(kk.y * dq, dq, acc);
        }
    }

    // odd-count tail (E is 8M here, but keep it general)
    for (int e = (npairs << 1) + tid; e < E; e += stride) {
        const int ia = idx[2 * e + 0];
        const int ib = idx[2 * e + 1];
        const float* xa = x + 3u * (unsigned)ia;
        const float* xb = x + 3u * (unsigned)ib;
        const float dx = xa[0] - xb[0];
        const float dy = xa[1] - xb[1];
        const float dz = xa[2] - xb[2];
        const float q  = fmaf(dz, dz, fmaf(dy, dy, dx * dx));
        const float le = l0[e];
        const float dq = fmaf(-le, le, q);
        acc = fmaf(ks[e] * dq, dq, acc);
    }

    __shared__ float smem[TPB / 32];
    const float bsum = block_sum(acc, smem);
    if (threadIdx.x == 0) partial[blockIdx.x] = bsum;
}

__global__ __launch_bounds__(TPB)
void stvk_final_kernel(const float* __restrict__ partial, int n,
                       float* __restrict__ out) {
    float acc = 0.0f;
    for (int i = threadIdx.x; i < n; i += TPB) acc += partial[i];

    __shared__ float smem[TPB / 32];
    const float total = block_sum(acc, smem);
    if (threadIdx.x == 0) out[0] = 0.5f * total;
}

extern "C" void kernel_launch(void* const* d_in, const int* in_sizes, int n_in,
                              void* d_out, int out_size, void* d_ws, size_t ws_size,
                              hipStream_t stream) {
    const float* x   = (const float*)d_in[0];   // [V,3] f32
    const float* l0  = (const float*)d_in[1];   // [E]   f32
    const float* ks  = (const float*)d_in[2];   // [E]   f32
    const int*   idx = (const int*)  d_in[3];   // [E,2] int32
    const int E = in_sizes[1];

    // Partial-sum buffer lives in d_ws; clamp block count to available scratch
    // (deterministic: ws_size is fixed across calls).
    int nblk = (int)(ws_size / sizeof(float));
    if (nblk > MAX_BLOCKS) nblk = MAX_BLOCKS;
    if (nblk < 1) nblk = 1;
    float* partial = (float*)d_ws;

    stvk_partial_kernel<<<nblk, TPB, 0, stream>>>(x, l0, ks, idx, partial, E);
    stvk_final_kernel<<<1, TPB, 0, stream>>>(partial, nblk, (float*)d_out);
}